// ODFormer_45140106281302
// MI455X (gfx1250) — compile-verified
//
#include <hip/hip_runtime.h>

// ODFormer periodic-causal attention for MI455X (gfx1250, wave32).
// All matmuls use V_WMMA_F32_16X16X4_F32 (native fp32 matrix op).
//
// Shapes: B=2, T=2048, D=512, H=8, HD=64.

#define B_  2
#define T_  2048
#define D_  512
#define H_  8
#define HD_ 64

typedef float v2f __attribute__((ext_vector_type(2)));
typedef float v8f __attribute__((ext_vector_type(8)));

// D(16x16,f32) = A(16x4,f32) * B(4x16,f32) + C
// A layout: lanes 0-15 -> M=lane, vgpr{0,1} = K{0,1}; lanes 16-31 -> K{2,3}
// B layout: lanes 0-15 -> N=lane, vgpr{0,1} = K{0,1}; lanes 16-31 -> K{2,3}
// C/D layout: lanes 0-15 -> N=lane, vgpr r = M=r; lanes 16-31 -> M=r+8
__device__ __forceinline__ v8f wmma4(v2f a, v2f b, v8f c) {
  return __builtin_amdgcn_wmma_f32_16x16x4_f32(
      /*neg_a=*/false, a, /*neg_b=*/false, b,
      /*c_mod=*/(short)0, c, /*reuse_a=*/false, /*reuse_b=*/false);
}

__device__ __forceinline__ float redmax16(float v) {
  #pragma unroll
  for (int m = 1; m < 16; m <<= 1) v = fmaxf(v, __shfl_xor(v, m, 32));
  return v;
}
__device__ __forceinline__ float redsum16(float v) {
  #pragma unroll
  for (int m = 1; m < 16; m <<= 1) v += __shfl_xor(v, m, 32);
  return v;
}

// ---------------------------------------------------------------------------
// Kernel 1: QKV projection. (BT=4096, 3D=1536) <- x(BT,512) @ w_qkv^T.
// Each wave: 16 rows x 64 cols, A fragment reused across 4 B tiles.
// Scatter into Q/K/V as (B,H,T,HD); Q pre-scaled by 1/sqrt(HD)=0.125.
// ---------------------------------------------------------------------------
__global__ __launch_bounds__(256) void qkv_gemm_kernel(
    const float* __restrict__ x, const float* __restrict__ w,
    const float* __restrict__ bias,
    float* __restrict__ Q, float* __restrict__ K, float* __restrict__ V) {
  const int lane = threadIdx.x & 31;
  const int wid  = threadIdx.x >> 5;
  const int cb   = blockIdx.x % 3;          // 3 col-blocks of 512
  const int rt   = blockIdx.x / 3;          // 256 row tiles of 16
  const int n0   = cb * 512 + wid * 64;
  const int row0 = rt * 16;

  const int l15   = lane & 15;
  const int koff  = (lane >> 4) * 2;
  const int mbase = (lane >> 4) * 8;

  v8f acc[4];
  #pragma unroll
  for (int j = 0; j < 4; ++j)
    #pragma unroll
    for (int r = 0; r < 8; ++r) acc[j][r] = 0.0f;

  const float* xrow = x + (size_t)(row0 + l15) * D_;
  for (int k0 = 0; k0 < D_; k0 += 4) {
    v2f a = *(const v2f*)(xrow + k0 + koff);
    #pragma unroll
    for (int j = 0; j < 4; ++j) {
      const float* wrow = w + (size_t)(n0 + j * 16 + l15) * D_;
      v2f b = *(const v2f*)(wrow + k0 + koff);
      acc[j] = wmma4(a, b, acc[j]);
    }
  }

  #pragma unroll
  for (int j = 0; j < 4; ++j) {
    const int oc    = n0 + j * 16 + l15;
    const float bv  = bias[oc];
    const int which = oc >> 9;              // 0=Q 1=K 2=V
    const int rem   = oc & 511;
    const int h     = rem >> 6;
    const int hd    = rem & 63;
    float* dst      = (which == 0) ? Q : (which == 1 ? K : V);
    const float scl = (which == 0) ? 0.125f : 1.0f;
    #pragma unroll
    for (int r = 0; r < 8; ++r) {
      const int tf = row0 + mbase + r;      // flat b*T + t
      const int b  = tf >> 11;              // / 2048
      const int t  = tf & 2047;
      dst[(((size_t)(b * H_ + h) * T_ + t) * HD_) + hd] = (acc[j][r] + bv) * scl;
    }
  }
}

// ---------------------------------------------------------------------------
// Kernel 2: flash attention with periodic-causal mask.
// One wave per (b,h,16-query tile). S=QK^T and O+=P*V via f32 WMMA.
// P goes D-layout -> A-layout through a per-wave LDS bounce (in-order DS).
// ---------------------------------------------------------------------------
__global__ __launch_bounds__(256) void pattn_kernel(
    const float* __restrict__ Q, const float* __restrict__ K,
    const float* __restrict__ V, const int* __restrict__ periods,
    float* __restrict__ AO) {
  __shared__ float ldsP[8][16 * 16];        // 1KB per wave

  const int lane = threadIdx.x & 31;
  const int wid  = threadIdx.x >> 5;
  const int gw   = blockIdx.x * 8 + wid;    // 0..2047
  const int qt   = gw & 127;                // query tile
  const int bh   = gw >> 7;                 // 0..15
  const int b    = bh >> 3;
  const int h    = bh & 7;

  const int q0    = qt * 16;
  const int l15   = lane & 15;
  const int koff  = (lane >> 4) * 2;
  const int mbase = (lane >> 4) * 8;

  int p = periods[bh];
  if (p < 1) p = 1;
  const unsigned up = (unsigned)p;

  const float* Qp = Q + (size_t)bh * T_ * HD_;
  const float* Kp = K + (size_t)bh * T_ * HD_;
  const float* Vp = V + (size_t)bh * T_ * HD_;

  // Q tile in A-layout: 16 k-slices of 4 (already scaled by 1/sqrt(HD)).
  v2f qreg[16];
  const float* qrow = Qp + (size_t)(q0 + l15) * HD_;
  #pragma unroll
  for (int s = 0; s < 16; ++s) qreg[s] = *(const v2f*)(qrow + 4 * s + koff);

  v8f O[4];
  float rmax[8], rsum[8];
  #pragma unroll
  for (int j = 0; j < 4; ++j)
    #pragma unroll
    for (int r = 0; r < 8; ++r) O[j][r] = 0.0f;
  #pragma unroll
  for (int r = 0; r < 8; ++r) { rmax[r] = -3.0e38f; rsum[r] = 0.0f; }

  float* myP = ldsP[wid];

  for (int kt = 0; kt <= qt; ++kt) {
    const int j0 = kt * 16;
    // Skip key tiles with no (i-j) % p == 0 pair in the 16x16 block.
    {
      const int dhi = q0 + 15 - j0;
      int dlo = q0 - (j0 + 15);
      if (dlo < 0) dlo = 0;
      if ((dhi / p) < ((dlo + p - 1) / p)) continue;
    }

    // S = Q * K^T  (16 WMMAs over HD=64)
    v8f S;
    #pragma unroll
    for (int r = 0; r < 8; ++r) S[r] = 0.0f;
    const float* krow = Kp + (size_t)(j0 + l15) * HD_;
    #pragma unroll
    for (int s = 0; s < 16; ++s) {
      v2f bfrag = *(const v2f*)(krow + 4 * s + koff);
      S = wmma4(qreg[s], bfrag, S);
    }

    // Mask, online softmax stats, rescale O.
    float pv[8];
    const int dbase = (q0 + mbase) - j0 - l15;  // diff for r = 0
    #pragma unroll
    for (int r = 0; r < 8; ++r) {
      const int diff = dbase + r;
      const bool valid = (diff >= 0) && ((unsigned)diff % up == 0u);
      const float sv = valid ? S[r] : -3.0e38f;
      const float tmax = redmax16(sv);
      const float mnew = fmaxf(rmax[r], tmax);
      const float alpha = __expf(rmax[r] - mnew);
      const float pe = valid ? __expf(sv - mnew) : 0.0f;
      const float tsum = redsum16(pe);
      rsum[r] = rsum[r] * alpha + tsum;
      rmax[r] = mnew;
      pv[r] = pe;
      #pragma unroll
      for (int j = 0; j < 4; ++j) O[j][r] *= alpha;
    }

    // P: D-layout -> A-layout via LDS (wave-private region, DS ops in-order).
    #pragma unroll
    for (int r = 0; r < 8; ++r) myP[(mbase + r) * 16 + l15] = pv[r];
    v2f pfrag[4];
    #pragma unroll
    for (int kk = 0; kk < 4; ++kk)
      pfrag[kk] = *(const v2f*)&myP[l15 * 16 + kk * 4 + koff];

    // O += P * V  (16 WMMAs: 4 key-slices x 4 d-tiles)
    #pragma unroll
    for (int kk = 0; kk < 4; ++kk) {
      const int kr = j0 + kk * 4 + koff;
      #pragma unroll
      for (int j = 0; j < 4; ++j) {
        v2f vb;
        vb[0] = Vp[(size_t)kr * HD_ + j * 16 + l15];
        vb[1] = Vp[(size_t)(kr + 1) * HD_ + j * 16 + l15];
        O[j] = wmma4(pfrag[kk], vb, O[j]);
      }
    }
  }

  // Normalize and store to (B,T,D) layout at column h*64.
  #pragma unroll
  for (int r = 0; r < 8; ++r) {
    const float inv = rsum[r] > 0.0f ? 1.0f / rsum[r] : 0.0f;
    const int t = q0 + mbase + r;
    float* orow = AO + ((size_t)b * T_ + t) * D_ + h * HD_;
    #pragma unroll
    for (int j = 0; j < 4; ++j) orow[j * 16 + l15] = O[j][r] * inv;
  }
}

// ---------------------------------------------------------------------------
// Kernel 3: output projection. out(BT,512) = ao(BT,512) @ w_out^T + b_out.
// ---------------------------------------------------------------------------
__global__ __launch_bounds__(256) void out_gemm_kernel(
    const float* __restrict__ ao, const float* __restrict__ w,
    const float* __restrict__ bias, float* __restrict__ out) {
  const int lane = threadIdx.x & 31;
  const int wid  = threadIdx.x >> 5;
  const int row0 = blockIdx.x * 16;
  const int n0   = wid * 64;

  const int l15   = lane & 15;
  const int koff  = (lane >> 4) * 2;
  const int mbase = (lane >> 4) * 8;

  v8f acc[4];
  #pragma unroll
  for (int j = 0; j < 4; ++j)
    #pragma unroll
    for (int r = 0; r < 8; ++r) acc[j][r] = 0.0f;

  const float* arow = ao + (size_t)(row0 + l15) * D_;
  for (int k0 = 0; k0 < D_; k0 += 4) {
    v2f a = *(const v2f*)(arow + k0 + koff);
    #pragma unroll
    for (int j = 0; j < 4; ++j) {
      const float* wrow = w + (size_t)(n0 + j * 16 + l15) * D_;
      v2f b = *(const v2f*)(wrow + k0 + koff);
      acc[j] = wmma4(a, b, acc[j]);
    }
  }

  #pragma unroll
  for (int j = 0; j < 4; ++j) {
    const int oc   = n0 + j * 16 + l15;
    const float bv = bias[oc];
    #pragma unroll
    for (int r = 0; r < 8; ++r)
      out[(size_t)(row0 + mbase + r) * D_ + oc] = acc[j][r] + bv;
  }
}

extern "C" void kernel_launch(void* const* d_in, const int* in_sizes, int n_in,
                              void* d_out, int out_size, void* d_ws, size_t ws_size,
                              hipStream_t stream) {
  (void)in_sizes; (void)n_in; (void)out_size; (void)ws_size;
  const float* x       = (const float*)d_in[0];
  const int*   periods = (const int*)d_in[1];
  const float* w_qkv   = (const float*)d_in[2];
  const float* b_qkv   = (const float*)d_in[3];
  const float* w_out   = (const float*)d_in[4];
  const float* b_out   = (const float*)d_in[5];

  float* ws = (float*)d_ws;
  const size_t nQ = (size_t)B_ * H_ * T_ * HD_;  // 2,097,152 floats each
  float* Q  = ws;
  float* K  = ws + nQ;
  float* V  = ws + 2 * nQ;
  float* AO = ws + 3 * nQ;                       // (B,T,D) pre-projection

  qkv_gemm_kernel<<<dim3(768), dim3(256), 0, stream>>>(x, w_qkv, b_qkv, Q, K, V);
  pattn_kernel<<<dim3(256), dim3(256), 0, stream>>>(Q, K, V, periods, AO);
  out_gemm_kernel<<<dim3(256), dim3(256), 0, stream>>>(AO, w_out, b_out,
                                                       (float*)d_out);
}